// ProjectionPointHead_68444598829771
// MI455X (gfx1250) — compile-verified
//
#include <hip/hip_runtime.h>
#include <cstdint>

// Problem constants (match reference)
#define B_ 16
#define C_ 128
#define N_ 16384
#define K_ 64
#define P_ 512            // points per chunk
#define NCHUNK_ (N_ / P_) // 32 chunks per cloud

// native clang vector type (required by __builtin_nontemporal_load)
typedef float v4f __attribute__((ext_vector_type(4)));

// ---------------------------------------------------------------------------
// Kernel 1: init workspace (B*K*C floats) to -inf. Must run every call since
// the harness does not re-poison between graph replays.
// ---------------------------------------------------------------------------
__global__ __launch_bounds__(256)
void seg_init_kernel(float* __restrict__ ws) {
    int g = blockIdx.x * 256 + threadIdx.x;
    if (g < B_ * K_ * C_) ws[g] = -__builtin_inff();
}

// ---------------------------------------------------------------------------
// Kernel 2: segmented max. One block = one (cloud b, 512-point chunk).
// 256 threads = 8 waves (wave32); warp -> channel, lane -> point (x4 float4).
//   - cluster ids staged to LDS with gfx1250 async copy (ASYNCcnt path),
//     then held in VGPRs and reused across all 16 channel iterations.
//   - feature stream read once -> non-temporal b128 loads (don't rinse L2).
//   - per-element reduce via native ds_max_num_f32 into LDS acc[c][k]
//     (layout chosen so LDS bank == k -> random-bank, no systematic conflict).
//   - block result combined into global ws via global_atomic_max_num_f32
//     (ws is 512 KB -> fully L2-resident on a 192 MB L2).
// ---------------------------------------------------------------------------
__global__ __launch_bounds__(256)
void seg_max_kernel(const float* __restrict__ pf,
                    const int*   __restrict__ cid,
                    float*       __restrict__ ws) {
    __shared__ float acc[C_ * K_];  // 32 KB, layout [c][k]
    __shared__ int   clds[P_];      // 2 KB staged cluster ids

    const int t    = threadIdx.x;
    const int lane = t & 31;
    const int warp = t >> 5;        // 0..7
    const int b    = blockIdx.y;    // cloud
    const int n0   = blockIdx.x * P_;

    // init LDS accumulator to -inf
    #pragma unroll
    for (int j = t; j < C_ * K_; j += 256) acc[j] = -__builtin_inff();

    // ---- stage 512 cluster ids: global -> LDS, async (ASYNCcnt path) ----
    {
        uint32_t lo  = (uint32_t)(uintptr_t)(&clds[t]);
        uint64_t ga  = (uint64_t)(uintptr_t)(cid + (size_t)b * N_ + n0 + t);
        asm volatile("global_load_async_to_lds_b32 %0, %1, off"
                     :: "v"(lo), "v"(ga) : "memory");
        uint32_t lo2 = lo + 256u * 4u;
        uint64_t ga2 = ga + 256u * 4u;
        asm volatile("global_load_async_to_lds_b32 %0, %1, off"
                     :: "v"(lo2), "v"(ga2) : "memory");
        asm volatile("s_wait_asynccnt 0" ::: "memory");
    }
    __syncthreads();

    // pull this lane's 16 cluster ids into registers (reused for 16 channels)
    int4 kk[4];
    {
        const int4* c4 = (const int4*)clds;
        #pragma unroll
        for (int p = 0; p < 4; ++p) kk[p] = c4[32 * p + lane];
    }

    // ---- main streaming reduce: 128 channels x 512 points ----
    for (int ci = 0; ci < C_ / 8; ++ci) {
        const int c = ci * 8 + warp;
        const v4f* src =
            (const v4f*)(pf + ((size_t)b * C_ + c) * N_ + n0);
        float* arow = acc + c * K_;
        #pragma unroll
        for (int p = 0; p < 4; ++p) {
            // read-once stream: non-temporal 512B/wave coalesced b128 load
            v4f  v = __builtin_nontemporal_load(&src[32 * p + lane]);
            int4 k = kk[p];
            __hip_atomic_fetch_max(&arow[k.x], v.x, __ATOMIC_RELAXED, __HIP_MEMORY_SCOPE_WORKGROUP);
            __hip_atomic_fetch_max(&arow[k.y], v.y, __ATOMIC_RELAXED, __HIP_MEMORY_SCOPE_WORKGROUP);
            __hip_atomic_fetch_max(&arow[k.z], v.z, __ATOMIC_RELAXED, __HIP_MEMORY_SCOPE_WORKGROUP);
            __hip_atomic_fetch_max(&arow[k.w], v.w, __ATOMIC_RELAXED, __HIP_MEMORY_SCOPE_WORKGROUP);
        }
    }
    __syncthreads();

    // ---- flush block partials: LDS acc[c][k] -> ws[b][k][c] (atomic max) ----
    float* wsb = ws + (size_t)b * K_ * C_;
    #pragma unroll
    for (int j = t; j < C_ * K_; j += 256) {
        int c = j >> 6;       // j / K_
        int k = j & (K_ - 1); // j % K_
        __hip_atomic_fetch_max(&wsb[(size_t)k * C_ + c], acc[j],
                               __ATOMIC_RELAXED, __HIP_MEMORY_SCOPE_AGENT);
    }
}

// ---------------------------------------------------------------------------
// Kernel 3: gather common cluster labels: out[b][i][c] = ws[b][common[b][i]][c]
// ---------------------------------------------------------------------------
__global__ __launch_bounds__(256)
void seg_gather_kernel(const float* __restrict__ ws,
                       const int*   __restrict__ common,
                       float*       __restrict__ out) {
    int g = blockIdx.x * 256 + threadIdx.x;
    if (g >= B_ * K_ * C_) return;
    int c = g & (C_ - 1);
    int i = (g >> 7) & (K_ - 1);
    int b = g >> 13;           // / (K_*C_)
    int k = common[b * K_ + i];
    float v = ws[((size_t)b * K_ + k) * C_ + c];   // ws: L2-resident
    __builtin_nontemporal_store(v, &out[g]);       // write-once output
}

extern "C" void kernel_launch(void* const* d_in, const int* in_sizes, int n_in,
                              void* d_out, int out_size, void* d_ws, size_t ws_size,
                              hipStream_t stream) {
    const float* pf     = (const float*)d_in[0]; // (B, C, N) f32
    const int*   cids   = (const int*)d_in[1];   // (B, N) i32
    const int*   common = (const int*)d_in[2];   // (B, K) i32
    float*       out    = (float*)d_out;         // (B*K, C) f32
    float*       ws     = (float*)d_ws;          // B*K*C f32 partial maxima

    const int total = B_ * K_ * C_; // 131072

    seg_init_kernel<<<(total + 255) / 256, 256, 0, stream>>>(ws);

    dim3 grid(NCHUNK_, B_);
    seg_max_kernel<<<grid, 256, 0, stream>>>(pf, cids, ws);

    seg_gather_kernel<<<(total + 255) / 256, 256, 0, stream>>>(ws, common, out);
}